// PartialDerivative_80530636800226
// MI455X (gfx1250) — compile-verified
//
#include <hip/hip_runtime.h>
#include <hip/hip_bf16.h>

// PartialDerivative along H: out[b,c,h,w] = x[b,c,h+1,w] - x[b,c,h,w]
// x: (16, 64, 512, 512) fp32  ->  out: (16, 64, 511, 512) fp32
//
// Pure streaming op (0.12 flop/byte): bound by HBM at ~2.07 GiB total traffic
// -> ~92 us floor on MI455X @ 23.3 TB/s. Kernel is organized purely around
// data movement: b128 coalesced loads, register row-reuse (each input row read
// ~1.12x instead of 2x), non-temporal b128 stores (write-once output, don't
// pollute the 192 MB L2), and global_prefetch_b8.

typedef float v4f __attribute__((ext_vector_type(4)));

// Geometry (hardcoded to the reference shapes).
static constexpr int PLANES   = 16 * 64;   // b*c = 1024
static constexpr int H        = 512;
static constexpr int W4       = 512 / 4;   // 128 float4 per row
static constexpr int HOUT     = H - 1;     // 511
static constexpr int RPB      = 8;         // output rows per block
static constexpr int CHUNKS   = (HOUT + RPB - 1) / RPB;  // 64 (last chunk = 7 rows)

__global__ __launch_bounds__(128) void diff_h_kernel(const v4f* __restrict__ x,
                                                     v4f* __restrict__ out) {
    const int plane = blockIdx.y;           // 0..1023  (b*c)
    const int h0    = blockIdx.x * RPB;     // 0, 8, ..., 504
    const int t     = threadIdx.x;          // 0..127, float4 index within row

    const v4f* __restrict__ xrow = x   + (size_t)plane * ((size_t)H    * W4) + (size_t)h0 * W4 + t;
    v4f*       __restrict__ orow = out + (size_t)plane * ((size_t)HOUT * W4) + (size_t)h0 * W4 + t;

    const int nrows = min(RPB, HOUT - h0);

    // Speculative prefetch of the next chunk's second input row, h0+RPB+1 (same plane).
    if (h0 + 2 * RPB <= HOUT) {
        __builtin_prefetch((const void*)(xrow + (size_t)(RPB + 1) * W4), 0, 0);
    }

    if (nrows == RPB) {
        // Fast path: issue all 9 row loads back-to-back for MLP, then 8 diffs.
        v4f r[RPB + 1];
#pragma unroll
        for (int i = 0; i <= RPB; ++i) {
            r[i] = xrow[(size_t)i * W4];
        }
#pragma unroll
        for (int i = 0; i < RPB; ++i) {
            v4f d = r[i + 1] - r[i];
            __builtin_nontemporal_store(d, &orow[(size_t)i * W4]);
        }
    } else {
        // Tail chunk (7 rows): rolling-register variant.
        v4f prev = xrow[0];
        for (int i = 0; i < nrows; ++i) {
            v4f cur = xrow[(size_t)(i + 1) * W4];
            v4f d   = cur - prev;
            __builtin_nontemporal_store(d, &orow[(size_t)i * W4]);
            prev = cur;
        }
    }
}

extern "C" void kernel_launch(void* const* d_in, const int* in_sizes, int n_in,
                              void* d_out, int out_size, void* d_ws, size_t ws_size,
                              hipStream_t stream) {
    (void)in_sizes; (void)n_in; (void)d_ws; (void)ws_size; (void)out_size;

    const v4f* x  = (const v4f*)d_in[0];
    v4f*       out = (v4f*)d_out;

    dim3 grid(CHUNKS, PLANES);  // (64, 1024)
    dim3 block(W4);             // 128 threads = 4 wave32 waves
    diff_h_kernel<<<grid, block, 0, stream>>>(x, out);
}